// PINNLoss_3719441679242
// MI455X (gfx1250) — compile-verified
//
#include <hip/hip_runtime.h>
#include <math.h>

// ---------------------------------------------------------------------------
// PINN loss on gfx1250 (MI455X).
// Forward Taylor-mode propagation of 6 streams (value, d/dx, d/dy, d/dt,
// d2/dx2, d2/dy2) through a 3->64->64->64->64->1 tanh MLP, with the 64x64
// hidden-layer matmuls batched across 32 points/block and executed on the
// WMMA pipe as V_WMMA_F32_16X16X4_F32 (fp32-exact, matches the fp32 ref).
// tanh is computed with the hardware exp pipe: tanh(x) = 1 - 2/(exp(2x)+1).
// ---------------------------------------------------------------------------

typedef __attribute__((ext_vector_type(2))) float v2f;
typedef __attribute__((ext_vector_type(8))) float v8f;

#define HID      64
#define PTS      32     // points per block
#define HSTRIDE  66     // LDS row stride (words) for activations: even -> 8B-aligned
                        // v2f A-operand loads, conflict-light ds_load_b64
#define WSTRIDE  72     // LDS row stride for weights: 2 rows -> +16 bank shift, so
                        // the two lane-halves of the B operand hit disjoint banks
#define NLAYERS  3      // hidden 64x64 layers

__device__ __forceinline__ float fast_tanh(float x)
{
    // v_exp_f32 + v_rcp_f32; saturates to +/-1 for large |x| (exp->inf or 0)
    const float e = __expf(2.0f * x);
    const float r = __builtin_amdgcn_rcpf(e + 1.0f);
    return fmaf(-2.0f, r, 1.0f);
}

// NS = 1: value-only forward pass (data loss branch)
// NS = 6: value + 3 first-derivative + 2 second-derivative streams (PDE branch)
template <int NS>
__global__ __launch_bounds__(256)
void pinn_mlp_kernel(const float* __restrict__ X,     // N x 3
                     const float* __restrict__ Y,     // N x 1 (NS==1 only)
                     const float* __restrict__ W0, const float* __restrict__ b0,
                     const float* __restrict__ W1, const float* __restrict__ b1,
                     const float* __restrict__ W2, const float* __restrict__ b2,
                     const float* __restrict__ W3, const float* __restrict__ b3,
                     const float* __restrict__ W4, const float* __restrict__ b4,
                     const float* __restrict__ log_alpha,
                     float* __restrict__ acc,         // acc[0]=data SSE, acc[1]=pde SSE
                     int N)
{
    extern __shared__ float smem[];
    float* sW   = smem;                          // 3 * 64 * WSTRIDE
    float* sb   = sW   + 3 * HID * WSTRIDE;      // 3 * 64
    float* sW4  = sb   + 3 * HID;                // 64
    float* sH   = sW4  + HID;                    // NS * PTS * HSTRIDE
    float* redp = sH   + NS * PTS * HSTRIDE;     // partials: 256 (NS==1) / 768 (NS==6)
    float* redr = redp + (NS == 6 ? 3 * 256 : 256); // PTS

    const int tid = threadIdx.x;

    // ---- preload hidden-layer weights into padded LDS; biases + W4 ----
    {
        const float* Wg[3] = { W1, W2, W3 };
        for (int idx = tid; idx < 3 * HID * HID; idx += 256) {
            const int l  = idx >> 12;
            const int rc = idx & 4095;
            const int r  = rc >> 6, c = rc & 63;
            sW[l * HID * WSTRIDE + r * WSTRIDE + c] = Wg[l][r * HID + c];
        }
        if (tid < HID) {
            sb[tid]           = b1[tid];
            sb[HID + tid]     = b2[tid];
            sb[2 * HID + tid] = b3[tid];
            sW4[tid]          = W4[tid];
        }
    }

    // ---- layer 0 seed: z = x@W0 + b0 (3->64), derivative seeds are W0 rows ----
    {
        const int m   = tid >> 3;       // point in block (0..31)
        const int grp = tid & 7;        // 8 features per thread
        const int g   = blockIdx.x * PTS + m;
        float x0 = 0.f, x1 = 0.f, x2 = 0.f;
        if (g < N) { x0 = X[3 * g]; x1 = X[3 * g + 1]; x2 = X[3 * g + 2]; }
#pragma unroll
        for (int j = 0; j < 8; ++j) {
            const int n = grp * 8 + j;
            const float w0 = W0[n], w1 = W0[HID + n], w2 = W0[2 * HID + n];
            const float z  = fmaf(x0, w0, fmaf(x1, w1, fmaf(x2, w2, b0[n])));
            const float hv = fast_tanh(z);
            const float s1 = 1.f - hv * hv;
            sH[0 * PTS * HSTRIDE + m * HSTRIDE + n] = hv;
            if (NS == 6) {
                const float gx = s1 * w0, gy = s1 * w1, gt = s1 * w2;
                sH[1 * PTS * HSTRIDE + m * HSTRIDE + n] = gx;
                sH[2 * PTS * HSTRIDE + m * HSTRIDE + n] = gy;
                sH[3 * PTS * HSTRIDE + m * HSTRIDE + n] = gt;
                // z'' = 0 at a linear layer: h'' = -2 h s (z')^2
                sH[4 * PTS * HSTRIDE + m * HSTRIDE + n] = -2.f * hv * gx * w0;
                sH[5 * PTS * HSTRIDE + m * HSTRIDE + n] = -2.f * hv * gy * w1;
            }
        }
    }
    __syncthreads();

    // ---- wave tiling: each wave owns one 16x16 (points x features) tile ----
    const int wv   = tid >> 5, lane = tid & 31;
    const int mt   = wv >> 2,  nt   = wv & 3;        // 2 M-tiles x 4 N-tiles
    const int half = lane >> 4, lq  = lane & 15;
    const int col  = nt * 16 + lq;                   // output feature (B/C column)
    const int mrow = mt * 16 + lq;                   // point row for A operand

    // ---- hidden layers: Z_s = H_s @ W_l, all 6 streams share B operand ----
    for (int l = 0; l < NLAYERS; ++l) {
        const float* Wl = sW + l * HID * WSTRIDE;
        v8f C[NS];
#pragma unroll
        for (int s = 0; s < NS; ++s) C[s] = {};

        for (int k0 = 0; k0 < HID; k0 += 4) {
            const int kk = k0 + 2 * half;            // A/B VGPR layout: lanes 16-31
                                                     // carry K = k0+2, k0+3
            v2f bv;
            bv.x = Wl[kk * WSTRIDE + col];
            bv.y = Wl[(kk + 1) * WSTRIDE + col];
#pragma unroll
            for (int s = 0; s < NS; ++s) {
                const v2f av =
                    *(const v2f*)&sH[s * PTS * HSTRIDE + mrow * HSTRIDE + kk];
                C[s] = __builtin_amdgcn_wmma_f32_16x16x4_f32(
                    /*neg_a=*/false, av, /*neg_b=*/false, bv,
                    /*c_mod=*/(short)0, C[s], /*reuse_a=*/false, /*reuse_b=*/false);
            }
        }
        __syncthreads();   // all waves done reading H before in-place overwrite

        const float bias = sb[l * HID + col];
#pragma unroll
        for (int i = 0; i < 8; ++i) {
            // C/D layout: VGPR i holds M=i (lanes 0-15) / M=i+8 (lanes 16-31)
            const int m  = mt * 16 + i + half * 8;
            const float zv = C[0][i] + bias;
            const float hv = fast_tanh(zv);
            const float s1 = 1.f - hv * hv;
            sH[0 * PTS * HSTRIDE + m * HSTRIDE + col] = hv;
            if (NS == 6) {
                const float zgx = C[1][i], zgy = C[2][i], zgt = C[3][i];
                const float zhx = C[4][i], zhy = C[5][i];
                const float hgx = s1 * zgx, hgy = s1 * zgy, hgt = s1 * zgt;
                sH[1 * PTS * HSTRIDE + m * HSTRIDE + col] = hgx;
                sH[2 * PTS * HSTRIDE + m * HSTRIDE + col] = hgy;
                sH[3 * PTS * HSTRIDE + m * HSTRIDE + col] = hgt;
                sH[4 * PTS * HSTRIDE + m * HSTRIDE + col] =
                    s1 * zhx - 2.f * hv * hgx * zgx;
                sH[5 * PTS * HSTRIDE + m * HSTRIDE + col] =
                    s1 * zhy - 2.f * hv * hgy * zgy;
            }
        }
        __syncthreads();
    }

    // ---- output layer 64 -> 1 (dot products), parallel partial reduction ----
    {
        const int m    = tid >> 3;      // point
        const int part = tid & 7;       // 8 features per thread
        const int kb   = part * 8;
        float p0 = 0.f, p1 = 0.f, p2 = 0.f;
#pragma unroll
        for (int j = 0; j < 8; ++j) {
            const int k   = kb + j;
            const float w = sW4[k];
            if (NS == 1) {
                p0 = fmaf(sH[m * HSTRIDE + k], w, p0);
            } else {
                p0 = fmaf(sH[3 * PTS * HSTRIDE + m * HSTRIDE + k], w, p0); // d/dt
                p1 = fmaf(sH[4 * PTS * HSTRIDE + m * HSTRIDE + k], w, p1); // d2/dx2
                p2 = fmaf(sH[5 * PTS * HSTRIDE + m * HSTRIDE + k], w, p2); // d2/dy2
            }
        }
        redp[tid] = p0;
        if (NS == 6) { redp[256 + tid] = p1; redp[512 + tid] = p2; }
    }
    __syncthreads();

    if (tid < PTS) {
        const int g = blockIdx.x * PTS + tid;
        float r2 = 0.f;
        if (g < N) {
            float a0 = 0.f, a1 = 0.f, a2 = 0.f;
#pragma unroll
            for (int p = 0; p < 8; ++p) {
                a0 += redp[tid * 8 + p];
                if (NS == 6) {
                    a1 += redp[256 + tid * 8 + p];
                    a2 += redp[512 + tid * 8 + p];
                }
            }
            if (NS == 1) {
                const float d = (a0 + b4[0]) - Y[g];
                r2 = d * d;
            } else {
                const float alpha = __expf(log_alpha[0]);
                const float res   = a0 - alpha * (a1 + a2);
                r2 = res * res;
            }
        }
        redr[tid] = r2;
    }
    __syncthreads();
    if (tid == 0) {
        float s = 0.f;
#pragma unroll
        for (int i = 0; i < PTS; ++i) s += redr[i];
        atomicAdd(&acc[NS == 1 ? 0 : 1], s);
    }
}

__global__ void pinn_finalize_kernel(const float* __restrict__ acc,
                                     const float* __restrict__ log_lambda,
                                     const float* __restrict__ log_alpha,
                                     float* __restrict__ out, float invN)
{
    if (threadIdx.x == 0 && blockIdx.x == 0) {
        const float dl  = acc[0] * invN;
        const float pl  = acc[1] * invN;
        const float lam = expf(log_lambda[0]);
        const float al  = expf(log_alpha[0]);
        out[0] = dl + lam * pl;   // total_loss
        out[1] = dl;              // data_loss
        out[2] = pl;              // pde_loss
        out[3] = al;              // alpha
        out[4] = lam;             // lam
    }
}

extern "C" void kernel_launch(void* const* d_in, const int* in_sizes, int n_in,
                              void* d_out, int out_size, void* d_ws, size_t ws_size,
                              hipStream_t stream)
{
    const float* x_data = (const float*)d_in[0];
    const float* y_data = (const float*)d_in[1];
    const float* x_phys = (const float*)d_in[2];
    const float* W0 = (const float*)d_in[3];  const float* b0 = (const float*)d_in[4];
    const float* W1 = (const float*)d_in[5];  const float* b1 = (const float*)d_in[6];
    const float* W2 = (const float*)d_in[7];  const float* b2 = (const float*)d_in[8];
    const float* W3 = (const float*)d_in[9];  const float* b3 = (const float*)d_in[10];
    const float* W4 = (const float*)d_in[11]; const float* b4 = (const float*)d_in[12];
    const float* log_lambda = (const float*)d_in[13];
    const float* log_alpha  = (const float*)d_in[14];

    const int N = in_sizes[0] / 3;
    float* acc = (float*)d_ws;
    hipMemsetAsync(d_ws, 0, 2 * sizeof(float), stream);  // graph-capturable memset node

    const int blocks = (N + PTS - 1) / PTS;
    const size_t ldsFixed = (size_t)(3 * HID * WSTRIDE + 3 * HID + HID) * sizeof(float);
    const size_t lds1 = ldsFixed +
        (size_t)(1 * PTS * HSTRIDE + 256 + PTS) * sizeof(float);
    const size_t lds6 = ldsFixed +
        (size_t)(6 * PTS * HSTRIDE + 3 * 256 + PTS) * sizeof(float);

    // data-loss branch: value stream only
    pinn_mlp_kernel<1><<<blocks, 256, lds1, stream>>>(
        x_data, y_data, W0, b0, W1, b1, W2, b2, W3, b3, W4, b4, log_alpha, acc, N);
    // PDE branch: 6 Taylor-mode streams
    pinn_mlp_kernel<6><<<blocks, 256, lds6, stream>>>(
        x_phys, nullptr, W0, b0, W1, b1, W2, b2, W3, b3, W4, b4, log_alpha, acc, N);

    pinn_finalize_kernel<<<1, 32, 0, stream>>>(
        acc, log_lambda, log_alpha, (float*)d_out, 1.0f / (float)N);
}